// NLCWNNet_DW_2370821947948
// MI455X (gfx1250) — compile-verified
//
#include <hip/hip_runtime.h>
#include <math.h>
#include <stdint.h>

// ---------------------------------------------------------------- constants
#define BATCH 64
#define CH    64
#define HH    112
#define WW    112
#define HWN   (HH*WW)        // 12544
#define NT    64             // pixels per attention chunk
#define NCHUNK (HWN/NT)      // 196
#define KSS   65             // ks LDS row stride (odd -> conflict-free A reads)
#define VSS   68             // vsT LDS row stride (4*l16 -> conflict-free)
#define EPSV  1e-5f

typedef __attribute__((ext_vector_type(2))) float v2f;
typedef __attribute__((ext_vector_type(4))) float f4;
typedef __attribute__((ext_vector_type(8))) float v8f;

__device__ __forceinline__ v8f wmma_f32_16x16x4(v2f a, v2f b, v8f c) {
    // 8 args: (neg_a, A, neg_b, B, c_mod, C, reuse_a, reuse_b)
    return __builtin_amdgcn_wmma_f32_16x16x4_f32(false, a, false, b, (short)0, c,
                                                 false, false);
}

// CDNA5 async global -> LDS copy (16B per lane), tracked by ASYNCcnt.
// vdst VGPR = LDS byte offset (low 32 bits of flat shared pointer).
__device__ __forceinline__ void async_copy_b128(void* lds_ptr, const void* gptr) {
    uint32_t lds_off = (uint32_t)(uintptr_t)lds_ptr;
    asm volatile("global_load_async_to_lds_b128 %0, %1, off"
                 :: "v"(lds_off), "v"(gptr)
                 : "memory");
}
__device__ __forceinline__ void wait_async0() {
    asm volatile("s_wait_asynccnt 0" ::: "memory");
}

// ---------------------------------------------------------------- zero ws
__global__ void zero_kernel(float* __restrict__ p, int n) {
    int i = blockIdx.x * blockDim.x + threadIdx.x;
    if (i < n) p[i] = 0.f;
}

// ---------------------------------------------------------------- fused attention -> context
// grid: (NCHUNK, BATCH), block: 256 threads (8 waves)
__global__ __launch_bounds__(256)
void attn_ctx_kernel(const float* __restrict__ x,
                     const float* __restrict__ Wv, const float* __restrict__ bv,
                     const float* __restrict__ Wm, const float* __restrict__ bm,
                     const float* __restrict__ bng, const float* __restrict__ bnb,
                     const float* __restrict__ bnm, const float* __restrict__ bnv,
                     float* __restrict__ ctx)
{
    __shared__ float xs [CH * NT];    // x tile   [c][n]
    __shared__ float ksl[CH * KSS];   // key tile [o][n] (padded)
    __shared__ float vsT[NT * VSS];   // value tile, transposed [n][o] (padded)

    const int b   = blockIdx.y;
    const int n0  = blockIdx.x * NT;
    const int tid = threadIdx.x;
    const int wid = tid >> 5;
    const int lane = tid & 31;
    const int l16 = lane & 15;
    const int lh  = lane >> 4;        // half-wave selects K-pair / +8 rows

    // ---- stage x tile into LDS via async copy (16B per lane)
    const float* gx = x + (size_t)b * CH * HWN + n0;
    for (int idx = tid; idx < CH * NT / 4; idx += 256) {
        int c  = idx >> 4;            // NT/4 = 16 float4 per channel row
        int n4 = idx & 15;
        async_copy_b128(&xs[c * NT + n4 * 4], gx + c * HWN + n4 * 4);
    }
    wait_async0();
    __syncthreads();

    // ---- both 1x1 convs as WMMA f32 GEMM tiles
    // tiles: 2 convs x 4 o-blocks x 4 n-blocks = 32 -> 4 per wave
    for (int t = wid; t < 32; t += 8) {
        const int isKey = t >> 4;
        const int rem   = t & 15;
        const int ob    = rem >> 2;
        const int nb    = rem & 3;
        const float* Wmat = isKey ? Wm : Wv;
        const int row = ob * 16 + l16;     // output channel (A row)
        const int col = nb * 16 + l16;     // pixel (B col)
        v8f acc = {0.f,0.f,0.f,0.f,0.f,0.f,0.f,0.f};
        #pragma unroll
        for (int ks = 0; ks < 16; ks++) {  // K = 64 channels, 4 per step
            int kb = 4 * ks + 2 * lh;
            v2f a;  a.x  = Wmat[row * CH + kb];
                    a.y  = Wmat[row * CH + kb + 1];
            v2f bb; bb.x = xs[kb * NT + col];
                    bb.y = xs[(kb + 1) * NT + col];
            acc = wmma_f32_16x16x4(a, bb, acc);
        }
        if (!isKey) {
            // value: store transposed [n][o] with bias
            #pragma unroll
            for (int r = 0; r < 8; r++) {
                int o = ob * 16 + r + 8 * lh;
                vsT[(nb * 16 + l16) * VSS + o] = acc[r] + bv[o];
            }
        } else {
            // key: fuse bias + BN affine, store [o][n]
            #pragma unroll
            for (int r = 0; r < 8; r++) {
                int o = ob * 16 + r + 8 * lh;
                float scl = bng[o] * rsqrtf(bnv[o] + EPSV);
                ksl[o * KSS + nb * 16 + l16] =
                    (acc[r] + bm[o] - bnm[o]) * scl + bnb[o];
            }
        }
    }
    __syncthreads();

    // ---- softmax over channels for each pixel (column of ksl)
    if (tid < NT) {
        float mx = -3.4e38f;
        for (int o = 0; o < CH; o++) mx = fmaxf(mx, ksl[o * KSS + tid]);
        float s = 0.f;
        for (int o = 0; o < CH; o++) {
            float e = __expf(ksl[o * KSS + tid] - mx);
            ksl[o * KSS + tid] = e;
            s += e;
        }
        float inv = 1.f / s;
        for (int o = 0; o < CH; o++) ksl[o * KSS + tid] *= inv;
    }
    __syncthreads();

    // ---- context GEMM: ctx[i][j] += sum_n softk[i][n] * value[n][j]
    // 16 tiles (4x4), 2 per wave; K = NT pixels
    for (int t = wid; t < 16; t += 8) {
        const int ib = t >> 2;
        const int jb = t & 3;
        const int irow = ib * 16 + l16;
        const int jcol = jb * 16 + l16;
        v8f acc = {0.f,0.f,0.f,0.f,0.f,0.f,0.f,0.f};
        #pragma unroll
        for (int kk = 0; kk < NT / 4; kk++) {
            int kb = 4 * kk + 2 * lh;
            v2f a;  a.x  = ksl[irow * KSS + kb];
                    a.y  = ksl[irow * KSS + kb + 1];
            v2f bb; bb.x = vsT[kb * VSS + jcol];
                    bb.y = vsT[(kb + 1) * VSS + jcol];
            acc = wmma_f32_16x16x4(a, bb, acc);
        }
        float* cb = ctx + (size_t)b * CH * CH;
        #pragma unroll
        for (int r = 0; r < 8; r++) {
            int i = ib * 16 + r + 8 * lh;
            atomicAdd(&cb[i * CH + jcol], acc[r]);
        }
    }
}

// ---------------------------------------------------------------- dynamic weights
// dynw[b][i][j] = bfc[i][j] + sum_m sigmoid(ctx[b][i][m]) * Wfc[i][j][m]
__global__ __launch_bounds__(256)
void dynw_kernel(const float* __restrict__ ctx, const float* __restrict__ Wfc,
                 const float* __restrict__ bfc, float* __restrict__ dynw)
{
    int g = blockIdx.x * 256 + threadIdx.x;   // g = b*64 + i
    if (g >= BATCH * CH) return;
    int bI = g >> 6, i = g & 63;
    const float* crow = ctx + (size_t)bI * CH * CH + (size_t)i * CH;
    const float* wrow = Wfc + (size_t)i * 9 * CH;
    float acc[9];
    #pragma unroll
    for (int j = 0; j < 9; j++) acc[j] = bfc[i * 9 + j];
    for (int m = 0; m < CH; m++) {
        float s = 1.f / (1.f + __expf(-crow[m]));
        #pragma unroll
        for (int j = 0; j < 9; j++) acc[j] += s * wrow[j * CH + m];
    }
    float* d = dynw + (size_t)g * 9;
    #pragma unroll
    for (int j = 0; j < 9; j++) d[j] = acc[j];
}

// ---------------------------------------------------------------- per-sample depthwise 3x3
// grid: B*C blocks, one 112x112 plane each; plane staged through LDS via
// CDNA5 async global->LDS copies (no VGPR round-trip).
__global__ __launch_bounds__(256)
void dwconv_kernel(const float* __restrict__ x, const float* __restrict__ dynw,
                   float* __restrict__ out)
{
    __shared__ float plane[HWN];      // 50176 B
    const int bc = blockIdx.x;
    const float* px = x + (size_t)bc * HWN;
    float* po = out + (size_t)bc * HWN;
    const int tid = threadIdx.x;

    for (int idx = tid; idx < HWN / 4; idx += 256)
        async_copy_b128(&plane[idx * 4], px + idx * 4);

    float w[9];
    #pragma unroll
    for (int j = 0; j < 9; j++) w[j] = dynw[bc * 9 + j];

    wait_async0();
    __syncthreads();

    for (int u = tid; u < HWN / 4; u += 256) {
        int y  = u / (WW / 4);
        int x0 = (u % (WW / 4)) * 4;
        float o[4] = {0.f, 0.f, 0.f, 0.f};
        #pragma unroll
        for (int ky = 0; ky < 3; ky++) {
            int yy = y + ky - 1;
            if (yy < 0 || yy >= HH) continue;
            const float* row = &plane[yy * WW];
            #pragma unroll
            for (int q = 0; q < 4; q++) {
                int xc = x0 + q;
                #pragma unroll
                for (int kx = 0; kx < 3; kx++) {
                    int xx = xc + kx - 1;
                    if (xx >= 0 && xx < WW) o[q] += w[ky * 3 + kx] * row[xx];
                }
            }
        }
        f4 ov; ov.x = o[0]; ov.y = o[1]; ov.z = o[2]; ov.w = o[3];
        *(f4*)(po + y * WW + x0) = ov;
    }
}

// ---------------------------------------------------------------- launch
extern "C" void kernel_launch(void* const* d_in, const int* in_sizes, int n_in,
                              void* d_out, int out_size, void* d_ws, size_t ws_size,
                              hipStream_t stream)
{
    const float* x    = (const float*)d_in[0];
    const float* Wv   = (const float*)d_in[1];
    const float* bv   = (const float*)d_in[2];
    const float* Wm   = (const float*)d_in[3];
    const float* bm   = (const float*)d_in[4];
    const float* bng  = (const float*)d_in[5];
    const float* bnb  = (const float*)d_in[6];
    const float* bnm  = (const float*)d_in[7];
    const float* bnv  = (const float*)d_in[8];
    const float* Wfc  = (const float*)d_in[9];
    const float* bfc  = (const float*)d_in[10];
    float* out = (float*)d_out;

    float* ctx  = (float*)d_ws;                       // B*C*C floats (1 MB)
    float* dynw = ctx + (size_t)BATCH * CH * CH;      // B*C*9 floats

    const int nctx = BATCH * CH * CH;
    zero_kernel<<<(nctx + 255) / 256, 256, 0, stream>>>(ctx, nctx);
    attn_ctx_kernel<<<dim3(NCHUNK, BATCH), 256, 0, stream>>>(
        x, Wv, bv, Wm, bm, bng, bnb, bnm, bnv, ctx);
    dynw_kernel<<<(BATCH * CH + 255) / 256, 256, 0, stream>>>(ctx, Wfc, bfc, dynw);
    dwconv_kernel<<<BATCH * CH, 256, 0, stream>>>(x, dynw, out);
}